// GlobalContextBlock_16647293240136
// MI455X (gfx1250) — compile-verified
//
#include <hip/hip_runtime.h>
#include <math.h>

// GlobalContextBlock, MI455X (gfx1250).
// Online-softmax fusion: x streamed from HBM only 2x (+1 write) instead of 3x:
//   k_fused : read x once; per (b, 512-l chunk) compute chunk logits, local
//             max/sum-exp, then re-sweep the chunk (L2-resident: ~80MB live
//             vs 192MB L2) for per-channel weighted partials.
//   k_combine: rescale partials by exp(m_i - M), normalize. (== global softmax)
//   k_mlp   : 16x256x64 MLP on one wave via V_WMMA_F32_16X16X4_F32.
//   k_out   : out = x + add  (read + write, float4 streamed).
// HBM traffic ~768 MB -> ~33us floor at 23.3 TB/s.

#define B_ 16
#define C_ 256
#define L_ 16384
#define M_ 64
#define CHL 512            // l-chunk per fused block
#define NCH (L_ / CHL)     // 32 chunks per batch
#define LN_EPS 1e-5f

typedef __attribute__((ext_vector_type(2))) float v2f;
typedef __attribute__((ext_vector_type(8))) float v8f;

// ---------------------------------------------------------------------------
// Kernel 1 (fused): per (b, chunk):
//   logits_l = sum_c x[b,c,l]*w_mask[c] + b_mask        (sweep 1, HBM read)
//   m = max_l logits, s = sum_l exp(logits-m), w_l = exp(logits-m)  (LDS)
//   pctx[c]  = sum_l x[b,c,l] * w_l                      (sweep 2, L2 re-read)
// grid = B*NCH, block = 256 (8 waves)
// ---------------------------------------------------------------------------
__global__ __launch_bounds__(256) void k_fused(const float* __restrict__ x,
                                               const float* __restrict__ w_mask,
                                               const float* __restrict__ b_mask,
                                               float* __restrict__ pm,
                                               float* __restrict__ ps,
                                               float* __restrict__ pctx) {
  const int b = blockIdx.x >> 5;       // NCH = 32 chunks per batch
  const int ch = blockIdx.x & 31;
  const int tid = threadIdx.x;
  const float* xb = x + (size_t)b * C_ * L_ + ch * CHL;

  __shared__ float sw[CHL];            // unnormalized weights exp(logit - m)
  __shared__ float red[256];

  // ---- sweep 1: logits for this chunk's 512 l (2 per thread, coalesced)
  float2 acc = {0.f, 0.f};
  const float* xt = xb + tid * 2;
  for (int c = 0; c < C_; ++c) {
    // CDNA5 speculative prefetch (global_prefetch_b8); OOB prefetch dropped.
    __builtin_prefetch(xt + (size_t)(c + 8) * L_, 0, 1);
    const float w = w_mask[c];
    const float2 v = *(const float2*)(xt + (size_t)c * L_);
    acc.x += v.x * w;
    acc.y += v.y * w;
  }
  const float bm = b_mask[0];
  acc.x += bm;
  acc.y += bm;

  // ---- local max over 512 logits
  red[tid] = fmaxf(acc.x, acc.y);
  __syncthreads();
  for (int s = 128; s > 0; s >>= 1) {
    if (tid < s) red[tid] = fmaxf(red[tid], red[tid + s]);
    __syncthreads();
  }
  const float m = red[0];
  __syncthreads();

  // ---- weights + local sum-exp
  const float e0 = expf(acc.x - m);
  const float e1 = expf(acc.y - m);
  sw[tid * 2 + 0] = e0;
  sw[tid * 2 + 1] = e1;
  red[tid] = e0 + e1;
  __syncthreads();
  for (int s = 128; s > 0; s >>= 1) {
    if (tid < s) red[tid] += red[tid + s];
    __syncthreads();
  }
  if (tid == 0) {
    pm[b * NCH + ch] = m;
    ps[b * NCH + ch] = red[0];
  }

  // ---- sweep 2: weighted per-channel partials (chunk now L2-resident).
  // wave w handles channels c = 8*i + w; lane owns l = lane*4 + j*128 so each
  // wave transaction is 512B contiguous and LDS weight reads are bank-spread.
  const int wave = tid >> 5;
  const int lane = tid & 31;
  float4 wv[4];
  for (int j = 0; j < 4; ++j)
    wv[j] = *(const float4*)(&sw[lane * 4 + j * 128]);
  float* pc = pctx + (size_t)(b * NCH + ch) * C_;
  for (int i = 0; i < C_ / 8; ++i) {
    const int c = i * 8 + wave;
    const float* xr = xb + (size_t)c * L_;
    float p = 0.f;
    for (int j = 0; j < 4; ++j) {
      const float4 v = *(const float4*)(xr + lane * 4 + j * 128);
      p += v.x * wv[j].x + v.y * wv[j].y + v.z * wv[j].z + v.w * wv[j].w;
    }
    for (int off = 16; off > 0; off >>= 1) p += __shfl_xor(p, off, 32);
    if (lane == 0) pc[c] = p;
  }
}

// ---------------------------------------------------------------------------
// Kernel 2: combine chunk partials with rescaling (exact global softmax):
//   M = max_i m_i ; denom = sum_i exp(m_i-M)*s_i
//   ctx[b,c] = sum_i exp(m_i-M)*pctx_i[c] / denom
// grid = B, block = 256 (one thread per channel)
// ---------------------------------------------------------------------------
__global__ __launch_bounds__(256) void k_combine(const float* __restrict__ pm,
                                                 const float* __restrict__ ps,
                                                 const float* __restrict__ pctx,
                                                 float* __restrict__ ctx) {
  const int b = blockIdx.x;
  const int tid = threadIdx.x;
  __shared__ float scale[NCH];
  __shared__ float sinv;

  if (tid < NCH) {                      // NCH==32: exactly wave 0, EXEC full
    const float mi = pm[b * NCH + tid];
    const float si = ps[b * NCH + tid];
    float M = mi;
    for (int off = 16; off > 0; off >>= 1) M = fmaxf(M, __shfl_xor(M, off, 32));
    const float sc = expf(mi - M);
    scale[tid] = sc;
    float d = sc * si;
    for (int off = 16; off > 0; off >>= 1) d += __shfl_xor(d, off, 32);
    if (tid == 0) sinv = 1.f / d;
  }
  __syncthreads();
  const float inv = sinv;
  float acc = 0.f;
  for (int i = 0; i < NCH; ++i)
    acc += scale[i] * pctx[(size_t)(b * NCH + i) * C_ + tid];
  ctx[b * C_ + tid] = acc * inv;
}

// ---------------------------------------------------------------------------
// Kernel 3: the MLP on one wave using V_WMMA_F32_16X16X4_F32 (M = batch = 16).
//   h(16x64)  = ctx(16x256) @ w1^T + b1  -> LayerNorm(64) -> ReLU
//   add(16x256) = h @ w2^T + b2
// ISA f32 layouts: A 16x4: VGPR v holds K=v (lanes 0-15) / K=v+2 (lanes 16-31),
// row M = lane&15. B 4x16 mirrored with N = lane&15. C/D: VGPR r holds
// row M=r (lanes 0-15) / M=r+8 (lanes 16-31), N = lane&15.
// ---------------------------------------------------------------------------
__global__ __launch_bounds__(256) void k_mlp(const float* __restrict__ ctx,
                                             const float* __restrict__ w1,
                                             const float* __restrict__ b1,
                                             const float* __restrict__ ln_g,
                                             const float* __restrict__ ln_b,
                                             const float* __restrict__ w2,
                                             const float* __restrict__ b2,
                                             float* __restrict__ add_out) {
  __shared__ float sctx[B_ * C_];  // 16 KB
  __shared__ float sh[B_ * M_];    // 4 KB
  const int tid = threadIdx.x;

  for (int i = tid; i < B_ * C_; i += 256) sctx[i] = ctx[i];
  __syncthreads();

  // ---- GEMM 1: h = ctx @ w1^T  (M=16 batch, N=64 mid, K=256) — wave 0 only,
  // all 32 lanes active so EXEC is all ones as WMMA requires.
  if (tid < 32) {
    const int lane = tid;
    const int half = lane >> 4;   // 0: K offset 0/1, 1: K offset 2/3
    const int lo = lane & 15;
    for (int n = 0; n < 4; ++n) {           // four 16-wide N tiles
      v8f acc = {};
      const float* brow = w1 + (size_t)(n * 16 + lo) * C_;  // w1[m_out, :]
      for (int k = 0; k < C_; k += 4) {
        v2f a, bv;
        const int kk = k + 2 * half;
        a.x = sctx[lo * C_ + kk];
        a.y = sctx[lo * C_ + kk + 1];
        bv.x = brow[kk];
        bv.y = brow[kk + 1];
        acc = __builtin_amdgcn_wmma_f32_16x16x4_f32(
            false, a, false, bv, (short)0, acc, false, false);
      }
      const int mcol = n * 16 + lo;
      const float bias = b1[mcol];
      for (int r = 0; r < 8; ++r) {
        const int brow_b = r + 8 * half;               // batch row
        sh[brow_b * M_ + mcol] = acc[r] + bias;
      }
    }
  }
  __syncthreads();

  // ---- LayerNorm over M=64 per batch row, then ReLU (16 rows, 1 thread each)
  if (tid < B_) {
    float mu = 0.f;
    for (int m = 0; m < M_; ++m) mu += sh[tid * M_ + m];
    mu *= (1.f / M_);
    float var = 0.f;
    for (int m = 0; m < M_; ++m) {
      const float d = sh[tid * M_ + m] - mu;
      var += d * d;
    }
    var *= (1.f / M_);
    const float rs = rsqrtf(var + LN_EPS);
    for (int m = 0; m < M_; ++m) {
      float v = (sh[tid * M_ + m] - mu) * rs * ln_g[m] + ln_b[m];
      sh[tid * M_ + m] = v > 0.f ? v : 0.f;
    }
  }
  __syncthreads();

  // ---- GEMM 2: add = h @ w2^T  (M=16 batch, N=256 channels, K=64)
  if (tid < 32) {
    const int lane = tid;
    const int half = lane >> 4;
    const int lo = lane & 15;
    for (int n = 0; n < 16; ++n) {          // sixteen 16-wide N tiles
      v8f acc = {};
      const float* brow = w2 + (size_t)(n * 16 + lo) * M_;  // w2[c, :]
      for (int k = 0; k < M_; k += 4) {
        v2f a, bv;
        const int kk = k + 2 * half;
        a.x = sh[lo * M_ + kk];
        a.y = sh[lo * M_ + kk + 1];
        bv.x = brow[kk];
        bv.y = brow[kk + 1];
        acc = __builtin_amdgcn_wmma_f32_16x16x4_f32(
            false, a, false, bv, (short)0, acc, false, false);
      }
      const int ccol = n * 16 + lo;
      const float bias = b2[ccol];
      for (int r = 0; r < 8; ++r) {
        const int brow_b = r + 8 * half;
        add_out[brow_b * C_ + ccol] = acc[r] + bias;
      }
    }
  }
}

// ---------------------------------------------------------------------------
// Kernel 4: out[b,c,l] = x[b,c,l] + add[b,c]   (float4 streamed)
// ---------------------------------------------------------------------------
__global__ __launch_bounds__(256) void k_out(const float* __restrict__ x,
                                             const float* __restrict__ addv,
                                             float* __restrict__ out) {
  const size_t gid = (size_t)blockIdx.x * 256 + threadIdx.x;
  const size_t idx = gid * 4;
  const int row = (int)(idx >> 14);   // idx / L_ -> b*C + c
  const float a = addv[row];
  const float4 v = *(const float4*)(x + idx);
  float4 o = {v.x + a, v.y + a, v.z + a, v.w + a};
  *(float4*)(out + idx) = o;
}

// ---------------------------------------------------------------------------
extern "C" void kernel_launch(void* const* d_in, const int* in_sizes, int n_in,
                              void* d_out, int out_size, void* d_ws, size_t ws_size,
                              hipStream_t stream) {
  (void)in_sizes; (void)n_in; (void)out_size; (void)ws_size;
  const float* x      = (const float*)d_in[0];
  const float* w_mask = (const float*)d_in[1];
  const float* b_mask = (const float*)d_in[2];
  const float* w1     = (const float*)d_in[3];
  const float* b1     = (const float*)d_in[4];
  const float* ln_g   = (const float*)d_in[5];
  const float* ln_b   = (const float*)d_in[6];
  const float* w2     = (const float*)d_in[7];
  const float* b2     = (const float*)d_in[8];
  float* out = (float*)d_out;

  // Workspace (floats): pm[512] | ps[512] | pctx[B*NCH*C = 131072] |
  //                     ctx[4096] | add[4096]  -> ~560 KB total.
  float* ws   = (float*)d_ws;
  float* pm   = ws;
  float* ps   = pm + B_ * NCH;
  float* pctx = ps + B_ * NCH;
  float* ctx  = pctx + (size_t)B_ * NCH * C_;
  float* addv = ctx + B_ * C_;

  k_fused  <<<B_ * NCH, 256, 0, stream>>>(x, w_mask, b_mask, pm, ps, pctx);
  k_combine<<<B_,       256, 0, stream>>>(pm, ps, pctx, ctx);
  k_mlp    <<<1,        256, 0, stream>>>(ctx, w1, b1, ln_g, ln_b, w2, b2, addv);
  k_out    <<<(B_ * C_ * L_ / 4) / 256, 256, 0, stream>>>(x, addv, out);
}